// SingleAttention_5179730559541
// MI455X (gfx1250) — compile-verified
//
#include <hip/hip_runtime.h>
#include <hip/hip_bf16.h>

#define BB 512
#define TT 512
#define DD 128
#define HH 128

typedef __attribute__((ext_vector_type(2))) float v2f;
typedef __attribute__((ext_vector_type(8))) float v8f;

// ---------------------------------------------------------------------------
// Kernel 1: q = last_visit @ Wt ; p = q @ Wx^T   (p[b,d] = sum_h q[b,h]*Wx[d,h])
// One block handles 16 batch rows. 8 waves, each owns a 16-wide N tile.
// Uses V_WMMA_F32_16X16X4_F32 (exact f32 FMA, matches reference numerics).
// ---------------------------------------------------------------------------
__global__ __launch_bounds__(256) void attn_qp(const float* __restrict__ input,
                                               const int*   __restrict__ mask,
                                               const float* __restrict__ Wt,
                                               const float* __restrict__ Wx,
                                               float*       __restrict__ p_ws) {
    __shared__ float lastv[16][DD];   // gathered last-visit rows
    __shared__ float qlds[16][HH];    // q tile staged between the two GEMMs
    __shared__ int   red[16][16];
    __shared__ int   lastidx[16];

    const int tid  = threadIdx.x;
    const int b0   = blockIdx.x * 16;
    const int lane = tid & 31;
    const int wave = tid >> 5;

    // ---- last_idx[b] = sum(mask[b,:]) - 1  (16 threads per row) ----
    {
        const int r = tid >> 4, c = tid & 15;
        const int* mrow = mask + (size_t)(b0 + r) * TT;
        int s = 0;
        #pragma unroll 4
        for (int j = 0; j < TT; j += 16) s += mrow[c + j];
        red[r][c] = s;
    }
    __syncthreads();
    if (tid < 16) {
        int tot = 0;
        #pragma unroll
        for (int j = 0; j < 16; ++j) tot += red[tid][j];
        int li = tot - 1;
        lastidx[tid] = li < 0 ? 0 : (li >= TT ? TT - 1 : li);
    }
    __syncthreads();

    // ---- gather last_visit rows into LDS ----
    for (int i = tid; i < 16 * DD; i += 256) {
        const int r = i >> 7, d = i & (DD - 1);
        lastv[r][d] = input[((size_t)(b0 + r) * TT + lastidx[r]) * DD + d];
    }
    __syncthreads();

    // WMMA f32 16x16x4 lane mapping
    const int m    = lane & 15;              // A row / also C col index source
    const int koff = (lane >> 4) * 2;        // lanes 16-31 hold K=2,3
    const int n    = lane & 15;              // B/C/D column within tile
    const int nb   = wave * 16;              // N tile base for this wave
    const int mrow0 = (lane >> 4) * 8;       // C/D: lanes 16-31 hold rows 8..15

    // ---- GEMM 1: q[m][n] = sum_k lastv[m][k] * Wt[k][n] ----
    v8f acc = {};
    for (int k = 0; k < DD; k += 4) {
        v2f a, b;
        a.x = lastv[m][k + koff];
        a.y = lastv[m][k + koff + 1];
        b.x = Wt[(size_t)(k + koff)     * HH + nb + n];
        b.y = Wt[(size_t)(k + koff + 1) * HH + nb + n];
        acc = __builtin_amdgcn_wmma_f32_16x16x4_f32(false, a, false, b,
                                                    (short)0, acc, false, false);
    }
    #pragma unroll
    for (int j = 0; j < 8; ++j) qlds[mrow0 + j][nb + n] = acc[j];
    __syncthreads();

    // ---- GEMM 2: p[m][d] = sum_h q[m][h] * Wx[d][h]  (B[k=h][n=d] = Wx[d*H+h]) ----
    v8f acc2 = {};
    for (int k = 0; k < HH; k += 4) {
        v2f a, b;
        a.x = qlds[m][k + koff];
        a.y = qlds[m][k + koff + 1];
        b.x = Wx[(size_t)(nb + n) * HH + k + koff];
        b.y = Wx[(size_t)(nb + n) * HH + k + koff + 1];
        acc2 = __builtin_amdgcn_wmma_f32_16x16x4_f32(false, a, false, b,
                                                     (short)0, acc2, false, false);
    }
    #pragma unroll
    for (int j = 0; j < 8; ++j)
        p_ws[(size_t)(b0 + mrow0 + j) * DD + nb + n] = acc2[j];
}

// ---------------------------------------------------------------------------
// Kernel 2: memory-bound fused pass. One block per batch row b.
// Pass 1: dot[t] = x[t,:] . p[b,:]  -> e[t] (sigmoid/log/decay) in LDS
// Softmax over t (block reduction), write a[b,:]
// Pass 2: v[b,:] = sum_t a[t] * x[t,:]  (input re-read hits L2: 134MB < 192MB)
// ---------------------------------------------------------------------------
__global__ __launch_bounds__(256) void attn_main(const float* __restrict__ input,
                                                 const int*   __restrict__ mask,
                                                 const float* __restrict__ p_ws,
                                                 const float* __restrict__ rate,
                                                 float* __restrict__ out_v,
                                                 float* __restrict__ out_a) {
    __shared__ float e_lds[TT];
    __shared__ float a_lds[TT];
    __shared__ float vred[8][DD];
    __shared__ float red[8];

    const int b    = blockIdx.x;
    const int tid  = threadIdx.x;
    const int lane = tid & 31;
    const int wave = tid >> 5;

    const float* xb = input + (size_t)b * TT * DD;

    // p fragment: 8 lanes per row; lane quad owns d0..d0+15
    const int quad = lane & 7;
    const int rg   = lane >> 3;      // row-in-wave 0..3
    const int d0   = quad * 16;
    const float4* pp = (const float4*)(p_ws + (size_t)b * DD + d0);
    const float4 p0 = pp[0], p1 = pp[1], p2 = pp[2], p3 = pp[3];

    const float rs = 1.0f / (1.0f + __expf(-rate[0]));   // sigmoid(rate)

    // ---- pass 1: dots + elementwise transform ----
    for (int it = 0; it < 16; ++it) {
        const int t = it * 32 + wave * 4 + rg;
        const float4* xr = (const float4*)(xb + (size_t)t * DD + d0);
        const float4 x0 = xr[0], x1 = xr[1], x2 = xr[2], x3 = xr[3];
        float part = x0.x * p0.x + x0.y * p0.y + x0.z * p0.z + x0.w * p0.w
                   + x1.x * p1.x + x1.y * p1.y + x1.z * p1.z + x1.w * p1.w
                   + x2.x * p2.x + x2.y * p2.y + x2.z * p2.z + x2.w * p2.w
                   + x3.x * p3.x + x3.y * p3.y + x3.z * p3.z + x3.w * p3.w;
        part += __shfl_xor(part, 1, 32);
        part += __shfl_xor(part, 2, 32);
        part += __shfl_xor(part, 4, 32);
        if (quad == 0) {
            const float sig   = 1.0f / (1.0f + __expf(-part));
            const float decay = (float)(TT - t);           // (t-1-i)+1
            const float denom = rs * (__logf(2.72f + (1.0f - sig)) * decay);
            float e = fmaxf(sig / denom, 0.0f);            // relu
            if (mask[(size_t)b * TT + t] == 0) e = -1e9f;
            e_lds[t] = e;
        }
    }
    __syncthreads();

    // ---- softmax over T=512 (each thread owns 2 entries) ----
    float mx = fmaxf(e_lds[tid], e_lds[tid + 256]);
    #pragma unroll
    for (int o = 16; o >= 1; o >>= 1) mx = fmaxf(mx, __shfl_xor(mx, o, 32));
    if (lane == 0) red[wave] = mx;
    __syncthreads();
    mx = red[0];
    #pragma unroll
    for (int w = 1; w < 8; ++w) mx = fmaxf(mx, red[w]);

    const float w0 = __expf(e_lds[tid] - mx);
    const float w1 = __expf(e_lds[tid + 256] - mx);
    float s = w0 + w1;
    #pragma unroll
    for (int o = 16; o >= 1; o >>= 1) s += __shfl_xor(s, o, 32);
    __syncthreads();                    // everyone done reading red[] (max)
    if (lane == 0) red[wave] = s;
    __syncthreads();
    s = 0.0f;
    #pragma unroll
    for (int w = 0; w < 8; ++w) s += red[w];
    const float inv = 1.0f / s;

    const float a0 = w0 * inv, a1 = w1 * inv;
    a_lds[tid]       = a0;
    a_lds[tid + 256] = a1;
    out_a[(size_t)b * TT + tid]       = a0;
    out_a[(size_t)b * TT + tid + 256] = a1;
    __syncthreads();

    // ---- pass 2: v = sum_t a[t]*x[t,:]  (coalesced float4 columns) ----
    const int col = lane;               // float4 column 0..31 -> d = col*4..col*4+3
    float4 acc = make_float4(0.f, 0.f, 0.f, 0.f);
    const float4* x4 = (const float4*)xb;    // row stride = 32 float4
    const int t0 = wave * 64;
    for (int t = t0; t < t0 + 64; ++t) {
        const float a = a_lds[t];
        const float4 x = x4[(size_t)t * 32 + col];
        acc.x += a * x.x; acc.y += a * x.y; acc.z += a * x.z; acc.w += a * x.w;
    }
    vred[wave][col * 4 + 0] = acc.x;
    vred[wave][col * 4 + 1] = acc.y;
    vred[wave][col * 4 + 2] = acc.z;
    vred[wave][col * 4 + 3] = acc.w;
    __syncthreads();
    if (tid < DD) {
        float v = 0.0f;
        #pragma unroll
        for (int w = 0; w < 8; ++w) v += vred[w][tid];
        out_v[(size_t)b * DD + tid] = v;
    }
}

extern "C" void kernel_launch(void* const* d_in, const int* in_sizes, int n_in,
                              void* d_out, int out_size, void* d_ws, size_t ws_size,
                              hipStream_t stream) {
    const float* input = (const float*)d_in[0];
    const int*   mask  = (const int*)d_in[1];
    const float* Wt    = (const float*)d_in[2];
    const float* Wx    = (const float*)d_in[3];
    const float* rate  = (const float*)d_in[4];

    float* p_ws  = (float*)d_ws;                 // [B,128] = 256 KB scratch
    float* out_v = (float*)d_out;                // [B,D]
    float* out_a = (float*)d_out + (size_t)BB * DD;  // [B,T]

    attn_qp  <<<BB / 16, 256, 0, stream>>>(input, mask, Wt, Wx, p_ws);
    attn_main<<<BB,      256, 0, stream>>>(input, mask, p_ws, rate, out_v, out_a);
}